// GraphFeature2_55370718380201
// MI455X (gfx1250) — compile-verified
//
#include <hip/hip_runtime.h>
#include <math.h>

typedef __attribute__((ext_vector_type(2))) float v2f;
typedef __attribute__((ext_vector_type(8))) float v8f;
typedef unsigned int u32x4 __attribute__((ext_vector_type(4)));
typedef unsigned int u32x8 __attribute__((ext_vector_type(8)));

// Native fp32 WMMA: D(16x16) = A(16x4) * B(4x16) + C
#define WMMA_F32(A, B, C) \
  __builtin_amdgcn_wmma_f32_16x16x4_f32(false, (A), false, (B), (short)0, (C), false, false)

constexpr int Bb = 8, Nn = 8192, Hh = 256, Ss = 32, Ff = 320; // F = H + A
constexpr int ROWS = Bb * Nn;   // 65536 rows of x / x_all
constexpr int TM = 32;          // M-tile per block (kernel 1)
// TDM pads 4 dwords after every 64 dwords -> effective LDS row stride:
//   320 + (320/64)*4 = 340 dwords.  340 % 64 == 20, and 20*dr % 64 == 0 only
//   at dr == 16 -> 16-lane A-fragment reads hit 16 distinct banks.
constexpr int LDSS = 340;

// ---------------------------------------------------------------------------
// Kernel 1: x_all = sigmoid(x*Wi^T + bi) * tanh(x*Wj^T + bj)
// grid = ROWS/TM blocks, 256 threads (8 waves). Wave w owns H-cols [32w, 32w+32).
// x tile (32 x 320 f32) staged into LDS by the Tensor Data Mover.
// ---------------------------------------------------------------------------
__global__ __launch_bounds__(256) void gate_gemm(
    const float* __restrict__ x, const float* __restrict__ Wi,
    const float* __restrict__ bi, const float* __restrict__ Wj,
    const float* __restrict__ bj, float* __restrict__ xall) {
  __shared__ float sx[TM * LDSS];

  const int tid = threadIdx.x;
  const int r0 = blockIdx.x * TM;
  const int w = tid >> 5, lane = tid & 31;
  const int lhalf = lane >> 4, l16 = lane & 15;
  const int n0 = w * 32;

  // ---- TDM: one tensor_load_to_lds for the whole 32x320 tile (wave 0 only).
  if (w == 0) {
    const unsigned long long ga =
        (unsigned long long)(uintptr_t)(x + (size_t)r0 * Ff);
    const unsigned lds_addr = (unsigned)(uintptr_t)(&sx[0]);  // low 32 = LDS offset

    // D# group 0: [1:0]=count=1, [63:32]=lds_addr, [120:64]=global_addr, [127:126]=type=2
    u32x4 g0;
    g0.x = 1u;
    g0.y = lds_addr;
    g0.z = (unsigned)(ga & 0xffffffffu);
    g0.w = (unsigned)((ga >> 32) & 0x01ffffffu) | (2u << 30);

    // D# group 1:
    //  [17:16] data_size=2 (4B)  [20] pad_enable=1
    //  [24:22] pad_interval=5 (64 dwords)  [31:25] pad_amount=3 (4 dwords)
    //  [79:48] tensor_dim0=320  [111:80] tensor_dim1=65536
    //  [127:112] tile_dim0=320  [143:128] tile_dim1=32  [159:144] tile_dim2=0
    //  [207:160] tensor_dim0_stride=320  [255:208] tensor_dim1_stride=0
    u32x8 g1;
    g1.s0 = (2u << 16) | (1u << 20) | (5u << 22) | (3u << 25);
    g1.s1 = (unsigned)(Ff & 0xffff) << 16;        // tensor_dim0[15:0] at bits 63:48
    g1.s2 = 0u;                                   // tensor_dim0[31:16]=0, tensor_dim1[15:0]=0
    g1.s3 = ((unsigned)Ff << 16) | 1u;            // tile_dim0=320 | tensor_dim1[31:16]=1
    g1.s4 = (unsigned)TM;                         // tile_dim1=32, tile_dim2=0
    g1.s5 = (unsigned)Ff;                         // tensor_dim0_stride[31:0]=320
    g1.s6 = 0u;
    g1.s7 = 0u;

    asm volatile("tensor_load_to_lds %0, %1" ::"s"(g0), "s"(g1) : "memory");
    __builtin_amdgcn_s_wait_tensorcnt(0);
  }

  // warm the W streams (gfx1250 global_prefetch_b8)
  __builtin_prefetch(Wi + (size_t)(n0 + l16) * Ff, 0, 1);
  __builtin_prefetch(Wj + (size_t)(n0 + l16) * Ff, 0, 1);

  __syncthreads();

  v8f acc_i[2][2] = {};
  v8f acc_j[2][2] = {};

  for (int k0 = 0; k0 < Ff; k0 += 4) {
    const int krow = k0 + lhalf * 2;
    const int kpad = krow + ((krow >> 6) << 2);   // TDM pad: +4 dwords per 64-dword chunk
    v2f a[2];
#pragma unroll
    for (int mt = 0; mt < 2; ++mt)
      a[mt] = *(const v2f*)(&sx[(mt * 16 + l16) * LDSS + kpad]);
#pragma unroll
    for (int nt = 0; nt < 2; ++nt) {
      const int col = n0 + nt * 16 + l16;
      const v2f bI = *(const v2f*)(Wi + (size_t)col * Ff + krow);
      const v2f bJ = *(const v2f*)(Wj + (size_t)col * Ff + krow);
#pragma unroll
      for (int mt = 0; mt < 2; ++mt) {
        acc_i[mt][nt] = WMMA_F32(a[mt], bI, acc_i[mt][nt]);
        acc_j[mt][nt] = WMMA_F32(a[mt], bJ, acc_j[mt][nt]);
      }
    }
  }

  // epilogue: bias + sigmoid*tanh, write x_all
#pragma unroll
  for (int nt = 0; nt < 2; ++nt) {
    const int col = n0 + nt * 16 + l16;
    const float bival = bi[col], bjval = bj[col];
#pragma unroll
    for (int mt = 0; mt < 2; ++mt) {
#pragma unroll
      for (int r = 0; r < 8; ++r) {
        const int row = r0 + mt * 16 + lhalf * 8 + r;
        const float si = 1.0f / (1.0f + expf(-(acc_i[mt][nt][r] + bival)));
        const float tj = tanhf(acc_j[mt][nt][r] + bjval);
        xall[(size_t)row * Hh + col] = si * tj;
      }
    }
  }
}

// ---------------------------------------------------------------------------
// Kernel 2: per-row L2 norm of x_all. One wave per row, grid = ROWS/8.
// ---------------------------------------------------------------------------
__global__ __launch_bounds__(256) void row_norm(const float* __restrict__ xall,
                                                float* __restrict__ nrm) {
  const int w = threadIdx.x >> 5, lane = threadIdx.x & 31;
  const int row = blockIdx.x * 8 + w;
  const float* p = xall + (size_t)row * Hh;
  float s = 0.f;
#pragma unroll
  for (int j = 0; j < 8; ++j) {
    const float v = p[lane + 32 * j];
    s += v * v;
  }
#pragma unroll
  for (int m = 16; m > 0; m >>= 1) s += __shfl_xor(s, m, 32);
  if (lane == 0) nrm[row] = sqrtf(s);
}

// ---------------------------------------------------------------------------
// Kernel 3: weight = mysoftmax(node_map * norm) over N. grid = B*S blocks.
// ---------------------------------------------------------------------------
__global__ __launch_bounds__(256) void masked_softmax(
    const float* __restrict__ node_map, const float* __restrict__ nrm,
    float* __restrict__ weight) {
  __shared__ float red[8];
  __shared__ float stot;
  const int tid = threadIdx.x;
  const int bs = blockIdx.x;  // b*S + s
  const int b = bs >> 5;
  const float* nm = node_map + (size_t)bs * Nn;
  const float* nb = nrm + (size_t)b * Nn;

  float s = 0.f;
  for (int n = tid; n < Nn; n += 256) {
    const float v = nm[n] * nb[n];
    s += (v > 0.f) ? expf(v) : 0.f;
  }
  const int w = tid >> 5, lane = tid & 31;
#pragma unroll
  for (int m = 16; m > 0; m >>= 1) s += __shfl_xor(s, m, 32);
  if (lane == 0) red[w] = s;
  __syncthreads();
  if (tid == 0) {
    float t = 0.f;
    for (int i = 0; i < 8; ++i) t += red[i];
    stot = t;
  }
  __syncthreads();
  const float part = stot;
  const float inv = (part > 0.f) ? 1.0f / part : 0.f;  // where(part>0, e/part, 0)
  for (int n = tid; n < Nn; n += 256) {
    const float v = nm[n] * nb[n];
    const float e = (v > 0.f) ? expf(v) : 0.f;
    weight[(size_t)bs * Nn + n] = e * inv;
  }
}

// ---------------------------------------------------------------------------
// Kernel 4: pooled[b,s,:] = tanh( weight[b,s,:] @ x_all[b,:,:] ) via WMMA.
// grid = B*8 blocks (32-col chunk of H each); K=8192 split across 8 waves,
// cross-wave LDS reduction.
// ---------------------------------------------------------------------------
__global__ __launch_bounds__(256) void pooled_gemm(
    const float* __restrict__ weight, const float* __restrict__ xall,
    float* __restrict__ outp) {
  __shared__ float alds[8 * 32 * 32];
  const int b = blockIdx.x >> 3, cg = blockIdx.x & 7;
  const int n0 = cg * 32;
  const int tid = threadIdx.x, w = tid >> 5, lane = tid & 31;
  const int lhalf = lane >> 4, l16 = lane & 15;
  const float* wgt = weight + (size_t)b * Ss * Nn;
  const float* xa = xall + (size_t)b * Nn * Hh;

  v8f acc[2][2] = {};
  for (int k0 = w * 4; k0 < Nn; k0 += 32) {
    const int krow = k0 + lhalf * 2;
    v2f a[2];
#pragma unroll
    for (int mt = 0; mt < 2; ++mt)
      a[mt] = *(const v2f*)(wgt + (size_t)(mt * 16 + l16) * Nn + krow);
#pragma unroll
    for (int nt = 0; nt < 2; ++nt) {
      const int col = n0 + nt * 16 + l16;
      v2f bv;
      bv[0] = xa[(size_t)krow * Hh + col];
      bv[1] = xa[(size_t)(krow + 1) * Hh + col];
#pragma unroll
      for (int mt = 0; mt < 2; ++mt) acc[mt][nt] = WMMA_F32(a[mt], bv, acc[mt][nt]);
    }
  }

#pragma unroll
  for (int mt = 0; mt < 2; ++mt)
#pragma unroll
    for (int nt = 0; nt < 2; ++nt)
#pragma unroll
      for (int r = 0; r < 8; ++r) {
        const int row = mt * 16 + lhalf * 8 + r;
        const int col = nt * 16 + l16;
        alds[w * 1024 + row * 32 + col] = acc[mt][nt][r];
      }
  __syncthreads();

  for (int idx = tid; idx < 1024; idx += 256) {
    float ssum = 0.f;
#pragma unroll
    for (int i = 0; i < 8; ++i) ssum += alds[i * 1024 + idx];
    const int row = idx >> 5, col = idx & 31;
    outp[((size_t)(b * Ss + row)) * Hh + n0 + col] = tanhf(ssum);
  }
}

// ---------------------------------------------------------------------------
extern "C" void kernel_launch(void* const* d_in, const int* in_sizes, int n_in,
                              void* d_out, int out_size, void* d_ws, size_t ws_size,
                              hipStream_t stream) {
  const float* x        = (const float*)d_in[0];  // [B, N, 320]
  const float* node_map = (const float*)d_in[1];  // [B, S, N]
  const float* Wi       = (const float*)d_in[2];  // [256, 320]
  const float* bi       = (const float*)d_in[3];  // [256]
  const float* Wj       = (const float*)d_in[4];  // [256, 320]
  const float* bj       = (const float*)d_in[5];  // [256]

  float* out        = (float*)d_out;
  float* pooled_out = out;                          // [B,S,H]  = 65536 floats
  float* weight_out = out + (size_t)Bb * Ss * Hh;   // [B,S,N,1] = 2097152 floats

  float* xall = (float*)d_ws;                       // 65536*256 floats (64 MB)
  float* nrm  = xall + (size_t)ROWS * Hh;           // 65536 floats

  gate_gemm<<<ROWS / TM, 256, 0, stream>>>(x, Wi, bi, Wj, bj, xall);
  row_norm<<<ROWS / 8, 256, 0, stream>>>(xall, nrm);
  masked_softmax<<<Bb * Ss, 256, 0, stream>>>(node_map, nrm, weight_out);
  pooled_gemm<<<Bb * 8, 256, 0, stream>>>(weight_out, xall, pooled_out);
}